// SinkhornDistance_3702261809881
// MI455X (gfx1250) — compile-verified
//
#include <hip/hip_runtime.h>
#include <math.h>

// MI455X / gfx1250, wave32. Sinkhorn distance between Gaussian mixtures.
//
//  1) prep_a / prep_b : traces, |mu|^2, log-weights, sqrt(cov_A) via
//     closed-form 3x3 eig + divided-difference polynomial (no cancellation).
//  2) cost_kernel     : one wave per 16x16 tile of C. 13 x
//     V_WMMA_F32_16X16X4_F32 per tile:
//       - 1 for the mu_A . mu_B^T dots (d=3 padded to K=4)
//       - 12 for the six entries of M = sqrt(A) B sqrt(A): each entry is a
//         K=6 dot of symmetrized sqrt(A)-features with covB's 6 unique
//         elements (B-symmetry compresses 9 dims -> 6, padded to K=8).
//     The accumulator layout hands each lane its 8 (i,j) pairs; only the
//     closed-form 3x3 eigensolve remains on the VALU.
//  3) 100 Sinkhorn iterations = 200 kernels (row/col streaming logsumexp).
//     C is 16 MB -> fully resident in the 192 MB L2: this phase is
//     L2-bandwidth/launch bound, ~16 MB per pass.
//  4) max + sum reductions -> scalar.

typedef __attribute__((ext_vector_type(2))) float v2f;
typedef __attribute__((ext_vector_type(8))) float v8f;

#define EIG_REG_F 1e-6f
#define LOG_EPS_F 1e-10f

static __device__ __forceinline__ float neg_inf() { return -__builtin_inff(); }

// Closed-form eigenvalues of a symmetric 3x3 (trigonometric method).
// Returns l1 >= l2 >= l3.
static __device__ __forceinline__ void eig3_sym(
    float a00, float a01, float a02, float a11, float a12, float a22,
    float& l1, float& l2, float& l3) {
  const float third = 0.33333333333333333f;
  float p1 = a01*a01 + a02*a02 + a12*a12;
  float q  = (a00 + a11 + a22) * third;
  float d0 = a00 - q, d1 = a11 - q, d2 = a22 - q;
  float p2 = d0*d0 + d1*d1 + d2*d2 + 2.0f*p1;
  float p  = sqrtf(p2 * (1.0f/6.0f));
  float pinv = (p > 1e-20f) ? (1.0f/p) : 0.0f;
  float b00 = d0*pinv, b11 = d1*pinv, b22 = d2*pinv;
  float b01 = a01*pinv, b02 = a02*pinv, b12 = a12*pinv;
  float detB = b00*(b11*b22 - b12*b12)
             - b01*(b01*b22 - b12*b02)
             + b02*(b01*b12 - b11*b02);
  float r = 0.5f*detB;
  r = fminf(1.0f, fmaxf(-1.0f, r));
  float phi = acosf(r) * third;
  float c0 = __cosf(phi);
  float c2 = __cosf(phi + 2.0943951023931953f);  // + 2*pi/3
  l1 = q + 2.0f*p*c0;
  l3 = q + 2.0f*p*c2;
  l2 = 3.0f*q - l1 - l3;
}

// ---------------------------------------------------------------------------
// Per-A precompute. sqrt(A) via Newton divided differences:
// sqrt(A) = s3*I + (A-l3 I)/(s2+s3) - (A-l3 I)(A-l2 I)/((s1+s2)(s2+s3)(s3+s1))
// Divided differences of sqrt collapse to 1/(si+sj): no cancellation, and the
// +0.05 I in setup keeps denominators >= 2*sqrt(0.05).
// ---------------------------------------------------------------------------
__global__ void prep_a_kernel(const float* __restrict__ muA,
                              const float* __restrict__ covA,
                              const float* __restrict__ wA,
                              float* __restrict__ sA6,
                              float* __restrict__ trA,
                              float* __restrict__ nA,
                              float* __restrict__ la,
                              float* __restrict__ lu,
                              int Na) {
  int a = blockIdx.x * blockDim.x + threadIdx.x;
  if (a >= Na) return;
  float x = muA[a*3+0], y = muA[a*3+1], z = muA[a*3+2];
  nA[a] = x*x + y*y + z*z;
  const float* Cv = covA + (size_t)a * 9;
  float a00 = Cv[0], a01 = Cv[1], a02 = Cv[2];
  float a11 = Cv[4], a12 = Cv[5], a22 = Cv[8];
  trA[a] = a00 + a11 + a22;
  float r00 = a00 + EIG_REG_F, r11 = a11 + EIG_REG_F, r22 = a22 + EIG_REG_F;
  float l1, l2, l3;
  eig3_sym(r00, a01, a02, r11, a12, r22, l1, l2, l3);
  float s1 = sqrtf(fmaxf(l1, 0.0f));
  float s2 = sqrtf(fmaxf(l2, 0.0f));
  float s3 = sqrtf(fmaxf(l3, 0.0f));
  float alpha = 1.0f / fmaxf(s2 + s3, 1e-12f);
  float beta  = 1.0f / fmaxf((s1 + s2) * (s2 + s3) * (s3 + s1), 1e-12f);
  // N1 = A - l3 I, N2 = A - l2 I; P = N1*N2 (symmetric: they commute)
  float n100 = r00 - l3, n111 = r11 - l3, n122 = r22 - l3;
  float n200 = r00 - l2, n211 = r11 - l2, n222 = r22 - l2;
  float p00 = n100*n200 + a01*a01 + a02*a02;
  float p01 = n100*a01 + a01*n211 + a02*a12;
  float p02 = n100*a02 + a01*a12 + a02*n222;
  float p11 = a01*a01 + n111*n211 + a12*a12;
  float p12 = a01*a02 + n111*a12 + a12*n222;
  float p22 = a02*a02 + a12*a12 + n122*n222;
  float* S = sA6 + (size_t)a * 6;
  S[0] = s3 + alpha*n100 - beta*p00;
  S[1] =      alpha*a01  - beta*p01;
  S[2] =      alpha*a02  - beta*p02;
  S[3] = s3 + alpha*n111 - beta*p11;
  S[4] =      alpha*a12  - beta*p12;
  S[5] = s3 + alpha*n122 - beta*p22;
  la[a] = logf(wA[a] + LOG_EPS_F);
  lu[a] = 0.0f;
}

__global__ void prep_b_kernel(const float* __restrict__ muB,
                              const float* __restrict__ covB,
                              const float* __restrict__ wB,
                              float* __restrict__ trB,
                              float* __restrict__ nB,
                              float* __restrict__ lb,
                              float* __restrict__ lv,
                              int Nb) {
  int b = blockIdx.x * blockDim.x + threadIdx.x;
  if (b >= Nb) return;
  float x = muB[b*3+0], y = muB[b*3+1], z = muB[b*3+2];
  nB[b] = x*x + y*y + z*z;
  const float* Cv = covB + (size_t)b * 9;
  trB[b] = Cv[0] + Cv[4] + Cv[8];
  lb[b] = logf(wB[b] + LOG_EPS_F);
  lv[b] = 0.0f;
}

// ---------------------------------------------------------------------------
// Cost matrix: one wave (32 lanes) per 16x16 tile; 13 WMMA per tile.
// f32 16x16x4 operand layout: per K-chunk of 4,
//   lanes 0-15 carry K={c0,c0+1} in v[0],v[1]; lanes 16-31 carry K={c0+2,c0+3}.
// D layout: VGPR r: lanes 0-15 -> M=r, lanes 16-31 -> M=r+8; N = lane&15.
// ---------------------------------------------------------------------------
__global__ void cost_kernel(const float* __restrict__ muA,
                            const float* __restrict__ muB,
                            const float* __restrict__ covB,
                            const float* __restrict__ sA6,
                            const float* __restrict__ trA,
                            const float* __restrict__ nA,
                            const float* __restrict__ trB,
                            const float* __restrict__ nB,
                            float* __restrict__ C,
                            int Na, int Nb) {
  const int lane = threadIdx.x;       // 0..31, one wave per block
  const int half = lane >> 4;         // 0 or 1
  const bool hi  = (half != 0);
  const int mrow = lane & 15;
  const int i0 = blockIdx.y * 16;
  const int j0 = blockIdx.x * 16;

  union U8 { v8f v; float f[8]; };
  v8f zero = {};

  // ---- WMMA #1: mean dots. A[m][k]=muA[i0+m][k], B[k][n]=muB[j0+n][k], K=4.
  const int k0 = half * 2;
  int ia  = i0 + mrow;
  int iac = (ia < Na) ? ia : (Na - 1);
  int jb  = j0 + mrow;
  int jbc = (jb < Nb) ? jb : (Nb - 1);
  v2f Aop, Bop;
  Aop[0] = ((k0 + 0) < 3) ? muA[iac*3 + k0 + 0] : 0.0f;
  Aop[1] = ((k0 + 1) < 3) ? muA[iac*3 + k0 + 1] : 0.0f;
  Bop[0] = ((k0 + 0) < 3) ? muB[jbc*3 + k0 + 0] : 0.0f;
  Bop[1] = ((k0 + 1) < 3) ? muB[jbc*3 + k0 + 1] : 0.0f;
  U8 dotAcc;
  dotAcc.v = __builtin_amdgcn_wmma_f32_16x16x4_f32(
      false, Aop, false, Bop, (short)0, zero, false, false);

  // ---- WMMA #2..#13: six entries of M = S B S as K=6 dots (padded to 8).
  // Feature dims kd=0..5 correspond to B's unique elements
  // (B00,B01,B02,B11,B12,B22); A-features symmetrize S[p,q]S[t,r].
  const float* Sp = sA6 + (size_t)iac * 6;
  float s0 = Sp[0], s1 = Sp[1], s2 = Sp[2], s3 = Sp[3], s4 = Sp[4], s5 = Sp[5];
  float Sf[3][3] = {{s0, s1, s2}, {s1, s3, s4}, {s2, s4, s5}};
  const float* Bp = covB + (size_t)jbc * 9;
  float b0 = Bp[0], b1 = Bp[1], b2 = Bp[2], b3 = Bp[4], b4 = Bp[5], b5 = Bp[8];

  // B operand (shared by all six entries): chunk0 kd{0..3}, chunk1 kd{4..7}.
  v2f G0, G1;
  G0[0] = hi ? b2 : b0;
  G0[1] = hi ? b3 : b1;
  G1[0] = hi ? 0.0f : b4;
  G1[1] = hi ? 0.0f : b5;

  U8 accM[6];
  constexpr int Pe[6] = {0, 0, 0, 1, 1, 2};
  constexpr int Re[6] = {0, 1, 2, 1, 2, 2};
#pragma unroll
  for (int e = 0; e < 6; ++e) {
    const int p = Pe[e], r = Re[e];
    float f0 = Sf[p][0]*Sf[0][r];
    float f1 = Sf[p][0]*Sf[1][r] + Sf[p][1]*Sf[0][r];
    float f2 = Sf[p][0]*Sf[2][r] + Sf[p][2]*Sf[0][r];
    float f3 = Sf[p][1]*Sf[1][r];
    float f4 = Sf[p][1]*Sf[2][r] + Sf[p][2]*Sf[1][r];
    float f5 = Sf[p][2]*Sf[2][r];
    v2f F0, F1;
    F0[0] = hi ? f2 : f0;
    F0[1] = hi ? f3 : f1;
    F1[0] = hi ? 0.0f : f4;
    F1[1] = hi ? 0.0f : f5;
    v8f acc = {};
    acc = __builtin_amdgcn_wmma_f32_16x16x4_f32(
        false, F0, false, G0, (short)0, acc, false, false);
    acc = __builtin_amdgcn_wmma_f32_16x16x4_f32(
        false, F1, false, G1, (short)0, acc, false, false);
    accM[e].v = acc;
  }

  // ---- per-lane epilogue: 8 (i,j) pairs straight from the accumulators.
  const int j = jb;
  const float tBj = trB[jbc];
  const float nBj = nB[jbc];
#pragma unroll
  for (int rr = 0; rr < 8; ++rr) {
    int i = i0 + rr + 8 * half;
    if (i >= Na || j >= Nb) continue;
    float m00 = accM[0].f[rr] + EIG_REG_F;
    float m01 = accM[1].f[rr];
    float m02 = accM[2].f[rr];
    float m11 = accM[3].f[rr] + EIG_REG_F;
    float m12 = accM[4].f[rr];
    float m22 = accM[5].f[rr] + EIG_REG_F;
    float l1, l2, l3;
    eig3_sym(m00, m01, m02, m11, m12, m22, l1, l2, l3);
    float trs = sqrtf(fmaxf(l1, 0.0f)) + sqrtf(fmaxf(l2, 0.0f)) +
                sqrtf(fmaxf(l3, 0.0f));
    float mean_term = nA[i] + nBj - 2.0f * dotAcc.f[rr];
    float cost = mean_term + trA[i] + tBj - 2.0f * trs;
    C[(size_t)i * Nb + j] = cost;
  }
}

// ---------------------------------------------------------------------------
// Sinkhorn: row update. One wave per row; streaming logsumexp + wave32 reduce.
// log_u[i] = log_a[i] - logsumexp_j(log_v[j] - C[i,j])
// ---------------------------------------------------------------------------
__global__ void row_update_kernel(const float* __restrict__ C,
                                  const float* __restrict__ lv,
                                  const float* __restrict__ la,
                                  float* __restrict__ lu,
                                  int Na, int Nb) {
  int wave = (int)((blockIdx.x * blockDim.x + threadIdx.x) >> 5);
  int lane = threadIdx.x & 31;
  if (wave >= Na) return;
  const float* row = C + (size_t)wave * Nb;
  float m = neg_inf(), s = 0.0f;
  for (int j = lane; j < Nb; j += 32) {
    float x = lv[j] - row[j];
    float nm = fmaxf(m, x);
    s = s * __expf(m - nm) + __expf(x - nm);
    m = nm;
  }
  for (int off = 16; off > 0; off >>= 1) {
    float om = __shfl_xor(m, off, 32);
    float os = __shfl_xor(s, off, 32);
    float nm = fmaxf(m, om);
    if (nm != neg_inf()) {
      s = s * __expf(m - nm) + os * __expf(om - nm);
      m = nm;
    }
  }
  if (lane == 0) lu[wave] = la[wave] - (m + __logf(s));
}

// ---------------------------------------------------------------------------
// Sinkhorn: column update. Block = 32 columns x 8 row-strips (coalesced).
// log_v[j] = log_b[j] - logsumexp_i(log_u[i] - C[i,j])
// ---------------------------------------------------------------------------
__global__ void col_update_kernel(const float* __restrict__ C,
                                  const float* __restrict__ lu,
                                  const float* __restrict__ lb,
                                  float* __restrict__ lv,
                                  int Na, int Nb) {
  __shared__ float sm[256];
  __shared__ float ss[256];
  int jj = threadIdx.x & 31;
  int rg = threadIdx.x >> 5;  // 0..7
  int j = blockIdx.x * 32 + jj;
  float m = neg_inf(), s = 0.0f;
  if (j < Nb) {
    for (int i = rg; i < Na; i += 8) {
      float x = lu[i] - C[(size_t)i * Nb + j];
      float nm = fmaxf(m, x);
      s = s * __expf(m - nm) + __expf(x - nm);
      m = nm;
    }
  }
  sm[threadIdx.x] = m;
  ss[threadIdx.x] = s;
  __syncthreads();
  if (threadIdx.x < 32 && j < Nb) {
    float M = sm[jj], S = ss[jj];
    for (int g = 1; g < 8; ++g) {
      float om = sm[g * 32 + jj], os = ss[g * 32 + jj];
      float nm = fmaxf(M, om);
      if (nm != neg_inf()) {
        S = S * __expf(M - nm) + os * __expf(om - nm);
        M = nm;
      }
    }
    lv[j] = lb[j] - (M + __logf(S));
  }
}

// ---------------------------------------------------------------------------
// Final reductions: global max of log_pi, then sum(exp(log_pi - max) * C).
// ---------------------------------------------------------------------------
__global__ void final_max_kernel(const float* __restrict__ C,
                                 const float* __restrict__ lu,
                                 const float* __restrict__ lv,
                                 float* __restrict__ partial,
                                 int Na, int Nb) {
  unsigned total = (unsigned)Na * (unsigned)Nb;
  unsigned stride = gridDim.x * blockDim.x;
  float m = neg_inf();
  for (unsigned idx = blockIdx.x * blockDim.x + threadIdx.x; idx < total;
       idx += stride) {
    unsigned i = idx / (unsigned)Nb;
    unsigned j = idx - i * (unsigned)Nb;
    float lp = lu[i] + lv[j] - C[idx];
    m = fmaxf(m, lp);
  }
  __shared__ float smem[256];
  smem[threadIdx.x] = m;
  __syncthreads();
  for (int s = 128; s > 0; s >>= 1) {
    if ((int)threadIdx.x < s)
      smem[threadIdx.x] = fmaxf(smem[threadIdx.x], smem[threadIdx.x + s]);
    __syncthreads();
  }
  if (threadIdx.x == 0) partial[blockIdx.x] = smem[0];
}

__global__ void reduce_max_kernel(const float* __restrict__ partial,
                                  float* __restrict__ gmax, int n) {
  __shared__ float smem[256];
  int t = threadIdx.x;
  smem[t] = (t < n) ? partial[t] : neg_inf();
  __syncthreads();
  for (int s = 128; s > 0; s >>= 1) {
    if (t < s) smem[t] = fmaxf(smem[t], smem[t + s]);
    __syncthreads();
  }
  if (t == 0) *gmax = smem[0];
}

__global__ void final_sum_kernel(const float* __restrict__ C,
                                 const float* __restrict__ lu,
                                 const float* __restrict__ lv,
                                 const float* __restrict__ gmax,
                                 float* __restrict__ partial,
                                 int Na, int Nb) {
  unsigned total = (unsigned)Na * (unsigned)Nb;
  unsigned stride = gridDim.x * blockDim.x;
  float gm = *gmax;
  float acc = 0.0f;
  for (unsigned idx = blockIdx.x * blockDim.x + threadIdx.x; idx < total;
       idx += stride) {
    unsigned i = idx / (unsigned)Nb;
    unsigned j = idx - i * (unsigned)Nb;
    float c = C[idx];
    float lp = lu[i] + lv[j] - c;
    acc += __expf(lp - gm) * c;
  }
  __shared__ float smem[256];
  smem[threadIdx.x] = acc;
  __syncthreads();
  for (int s = 128; s > 0; s >>= 1) {
    if ((int)threadIdx.x < s) smem[threadIdx.x] += smem[threadIdx.x + s];
    __syncthreads();
  }
  if (threadIdx.x == 0) partial[blockIdx.x] = smem[0];
}

__global__ void reduce_sum_kernel(const float* __restrict__ partial,
                                  float* __restrict__ out, int n) {
  __shared__ float smem[256];
  int t = threadIdx.x;
  smem[t] = (t < n) ? partial[t] : 0.0f;
  __syncthreads();
  for (int s = 128; s > 0; s >>= 1) {
    if (t < s) smem[t] += smem[t + s];
    __syncthreads();
  }
  if (t == 0) out[0] = smem[0];
}

// ---------------------------------------------------------------------------
extern "C" void kernel_launch(void* const* d_in, const int* in_sizes, int n_in,
                              void* d_out, int out_size, void* d_ws,
                              size_t ws_size, hipStream_t stream) {
  const float* muA  = (const float*)d_in[0];
  const float* covA = (const float*)d_in[1];
  const float* wA   = (const float*)d_in[2];
  const float* muB  = (const float*)d_in[3];
  const float* covB = (const float*)d_in[4];
  const float* wB   = (const float*)d_in[5];
  const int Na = in_sizes[2];
  const int Nb = in_sizes[5];

  float* w = (float*)d_ws;
  size_t off = 0;
  float* C    = w + off; off += (size_t)Na * Nb;  // 16 MB: L2-resident
  float* sA6  = w + off; off += (size_t)Na * 6;
  float* trA  = w + off; off += Na;
  float* nA   = w + off; off += Na;
  float* la   = w + off; off += Na;
  float* lu   = w + off; off += Na;
  float* trB  = w + off; off += Nb;
  float* nB   = w + off; off += Nb;
  float* lb   = w + off; off += Nb;
  float* lv   = w + off; off += Nb;
  float* part = w + off; off += 256;
  float* gmax = w + off; off += 1;

  prep_a_kernel<<<(Na + 255) / 256, 256, 0, stream>>>(muA, covA, wA, sA6, trA,
                                                      nA, la, lu, Na);
  prep_b_kernel<<<(Nb + 255) / 256, 256, 0, stream>>>(muB, covB, wB, trB, nB,
                                                      lb, lv, Nb);
  dim3 cgrid((Nb + 15) / 16, (Na + 15) / 16);
  cost_kernel<<<cgrid, 32, 0, stream>>>(muA, muB, covB, sA6, trA, nA, trB, nB,
                                        C, Na, Nb);

  int rowBlocks = (Na + 7) / 8;    // 8 waves/block, one wave per row
  int colBlocks = (Nb + 31) / 32;  // 32 columns per block
  for (int it = 0; it < 100; ++it) {
    row_update_kernel<<<rowBlocks, 256, 0, stream>>>(C, lv, la, lu, Na, Nb);
    col_update_kernel<<<colBlocks, 256, 0, stream>>>(C, lu, lb, lv, Na, Nb);
  }

  final_max_kernel<<<256, 256, 0, stream>>>(C, lu, lv, part, Na, Nb);
  reduce_max_kernel<<<1, 256, 0, stream>>>(part, gmax, 256);
  final_sum_kernel<<<256, 256, 0, stream>>>(C, lu, lv, gmax, part, Na, Nb);
  reduce_sum_kernel<<<1, 256, 0, stream>>>(part, (float*)d_out, 256);
}